// Mamba_Fusion_Full_78546361909287
// MI455X (gfx1250) — compile-verified
//
#include <hip/hip_runtime.h>
#include <hip/hip_bf16.h>
#include <math.h>

// ---------------------------------------------------------------------------
// Mamba tri-branch fusion for MI455X (gfx1250), wave32, WMMA f32 16x16x4.
// Only the last timestep feeds the classifier => scan carries state only,
// out_proj/fusion/classifier run on 8 tokens total.
//   P0: transpose in_proj x-half -> ipT (k,n)
//   P1: BT = concat( (dt_w @ x_proj[:8])^T , x_proj[8:40]^T )  (256 x 288)
//       -> single unconditional B stream for the delta/B/C GEMM
//   K1: emb GEMM -> LayerNorm -> in_proj GEMM  (WMMA, fused via LDS)
//   K1b: z at t=L-1 only
//   K2: causal depthwise conv4 + SiLU
//   K3: delta=softplus(..), emits interleaved (delta, delta*u); B/C GEMM (WMMA)
//   K4: selective scan, final state only: 1 x b64 + 1 x b128-LDS per step
//   K5: per-branch out_proj at last token
//   K6: fusion GEMM + LayerNorm + classifier
// ---------------------------------------------------------------------------

typedef __attribute__((ext_vector_type(2))) float v2f;
typedef __attribute__((ext_vector_type(4))) float v4f;
typedef __attribute__((ext_vector_type(8))) float v8f;

#define BATCH 8
#define SEQ   2048
#define DM    128      // d_model
#define DI    256      // d_inner
#define DS    16       // d_state
#define BL    (BATCH*SEQ)
#define NB    288      // BT width: 256 delta cols + 32 B/C cols

__device__ __forceinline__ v8f wmma_f32_16x16x4(v2f a, v2f b, v8f c) {
    return __builtin_amdgcn_wmma_f32_16x16x4_f32(false, a, false, b, (short)0, c,
                                                 false, false);
}

// ---------------- P0: ipT[k][n] = in_proj[n][k], n<256, k<128 ---------------
__global__ void transpose_ip_kernel(const float* __restrict__ in_proj,
                                    float* __restrict__ ipT) {
    const int k = blockIdx.x;    // 128
    const int n = threadIdx.x;   // 256
    ipT[k * DI + n] = in_proj[(size_t)n * DM + k];
}

// ---------------- P1: BT (256 x 288): [Wd^T | xp^T] -------------------------
__global__ void wprep_kernel(const float* __restrict__ dt_w,
                             const float* __restrict__ x_proj,
                             float* __restrict__ BT) {   // (256,288) (k, col)
    const int k = blockIdx.x;    // 256
    const int t = threadIdx.x;   // 288
    float s;
    if (t < 256) {               // W_delta^T = (dt_w @ x_proj[:8])^T
        s = 0.f;
#pragma unroll
        for (int r = 0; r < 8; ++r) s += dt_w[t * 8 + r] * x_proj[r * DI + k];
    } else {                     // x_proj rows 8..39 transposed
        s = x_proj[(size_t)(8 + (t - 256)) * DI + k];
    }
    BT[k * NB + t] = s;
}

// ---------------- K1: emb GEMM -> LN -> in_proj GEMM (x half) ---------------
// block = 256 threads (8 waves), 32 tokens per block, grid = BL/32
template <int K1>
__global__ void branch_front_kernel(const float* __restrict__ x,       // (BL,K1)
                                    const float* __restrict__ W_emb,   // (K1,128)
                                    const float* __restrict__ b_emb,
                                    const float* __restrict__ ln_g,
                                    const float* __restrict__ ln_b,
                                    const float* __restrict__ ipT,     // (128,256)
                                    float* __restrict__ x_in,          // (BL,256)
                                    float* __restrict__ h_last,        // (8,128)
                                    float* __restrict__ hn_last) {     // (8,128)
    __shared__ float sh[32][134];           // 134: 8B-aligned rows, 6 mod 64 banks
    const int tid  = threadIdx.x;
    const int wave = tid >> 5;
    const int lane = tid & 31;
    const int half = lane >> 4;
    const int r    = lane & 15;
    const int tok0 = blockIdx.x * 32;

    // ---- stage 1: h = x @ W_emb + b_emb (2 row-tiles x 8 col-tiles) ----
#pragma unroll
    for (int tno = 0; tno < 2; ++tno) {
        const int tile = wave + tno * 8;    // 0..15
        const int rt = tile >> 3;
        const int ct = tile & 7;
        const int n  = ct * 16 + r;
        const float* arow = x + (size_t)(tok0 + rt * 16 + r) * K1 + 2 * half;
        v8f acc = {0.f, 0.f, 0.f, 0.f, 0.f, 0.f, 0.f, 0.f};
#pragma unroll
        for (int k0 = 0; k0 < K1; k0 += 4) {
            const v2f a = *(const v2f*)(arow + k0);          // one b64
            v2f b;
            b[0] = W_emb[(size_t)(k0 + 2 * half)     * DM + n];
            b[1] = W_emb[(size_t)(k0 + 2 * half + 1) * DM + n];
            acc = wmma_f32_16x16x4(a, b, acc);
        }
#pragma unroll
        for (int v = 0; v < 8; ++v) {
            const int m  = rt * 16 + v + half * 8;
            const int nn = ct * 16 + r;
            sh[m][nn] = acc[v] + b_emb[nn];
        }
    }
    __syncthreads();

    // ---- LayerNorm over 128, 8 threads/row, shfl reduce ----
    {
        const int row = tid >> 3;
        const int sub = tid & 7;
        float s = 0.f, s2 = 0.f;
#pragma unroll
        for (int i = 0; i < 16; ++i) {
            const float v = sh[row][sub * 16 + i];
            s += v; s2 += v * v;
        }
#pragma unroll
        for (int m = 1; m < 8; m <<= 1) {
            s  += __shfl_xor(s,  m, 32);
            s2 += __shfl_xor(s2, m, 32);
        }
        const float mean = s * (1.f / 128.f);
        const float var  = s2 * (1.f / 128.f) - mean * mean;
        const float rstd = rsqrtf(var + 1e-5f);
        const int gt = tok0 + row;
        const bool is_last = ((gt & (SEQ - 1)) == (SEQ - 1));
        const int bidx = gt >> 11;
#pragma unroll
        for (int i = 0; i < 16; ++i) {
            const int nn = sub * 16 + i;
            const float v = sh[row][nn];
            if (is_last) h_last[bidx * DM + nn] = v;
            const float hn = (v - mean) * rstd * ln_g[nn] + ln_b[nn];
            if (is_last) hn_last[bidx * DM + nn] = hn;
            sh[row][nn] = hn;
        }
    }
    __syncthreads();

    // ---- stage 2: x_in = hn @ ipT (K=128, 2 rt x 16 ct) ----
#pragma unroll
    for (int tno = 0; tno < 4; ++tno) {
        const int tile = wave + tno * 8;    // 0..31
        const int ct = tile & 15;
        const int rt = tile >> 4;
        const int n  = ct * 16 + r;
        v8f acc = {0.f, 0.f, 0.f, 0.f, 0.f, 0.f, 0.f, 0.f};
#pragma unroll 8
        for (int k0 = 0; k0 < DM; k0 += 4) {
            const v2f a = *(const v2f*)(&sh[rt * 16 + r][k0 + 2 * half]); // ds b64
            v2f b;
            b[0] = ipT[(size_t)(k0 + 2 * half)     * DI + n];  // lane-contiguous
            b[1] = ipT[(size_t)(k0 + 2 * half + 1) * DI + n];
            acc = wmma_f32_16x16x4(a, b, acc);
        }
#pragma unroll
        for (int v = 0; v < 8; ++v) {
            const int m = rt * 16 + v + half * 8;
            x_in[(size_t)(tok0 + m) * DI + n] = acc[v];
        }
    }
}

// ---------------- K1b: z at last token only ---------------------------------
__global__ void z_last_kernel(const float* __restrict__ hn_last,
                              const float* __restrict__ in_proj,
                              float* __restrict__ z_last) {
    const int n = threadIdx.x;   // 256
    const int b = blockIdx.x;    // 8
    const float* w = in_proj + (size_t)(DI + n) * DM;
    const float* h = hn_last + b * DM;
    float s = 0.f;
#pragma unroll 8
    for (int k = 0; k < DM; ++k) s += h[k] * w[k];
    z_last[b * DI + n] = s;
}

// ---------------- K2: causal depthwise conv (k=4) + SiLU --------------------
__global__ void conv_silu_kernel(const float* __restrict__ x_in,
                                 const float* __restrict__ conv_w,  // (256,1,4)
                                 const float* __restrict__ conv_b,
                                 float* __restrict__ x_act) {
    const int idx = blockIdx.x * blockDim.x + threadIdx.x;  // BL*256
    const int c = idx & 255;
    const int tg = idx >> 8;          // b*SEQ + t
    const int t = tg & (SEQ - 1);
    float acc = conv_b[c];
#pragma unroll
    for (int j = 0; j < 4; ++j) {
        const int tt = t - 3 + j;
        if (tt >= 0)
            acc += conv_w[c * 4 + j] * x_in[(size_t)(tg - 3 + j) * DI + c];
    }
    x_act[idx] = acc / (1.f + __expf(-acc));  // silu
}

// ---------------- K3: delta (softplus) + (delta,delta*u) pairs; B/C GEMM ----
// block = 256 threads, 32 tokens; N = 288 (one BT stream) => 36 tiles
__global__ void dbc_kernel(const float* __restrict__ x_act,
                           const float* __restrict__ BT,      // (256,288)
                           const float* __restrict__ dt_b,    // (256,)
                           float* __restrict__ dd,            // (BL,256,2)
                           float* __restrict__ Bm,            // (BL,16)
                           float* __restrict__ Cm) {          // (BL,16)
    __shared__ float sa[32][262];           // 8B-aligned rows, 6 mod 64 banks
    const int tid  = threadIdx.x;
    const int tok0 = blockIdx.x * 32;
    for (int i = tid; i < 32 * DI / 2; i += 256) {     // v2f staging
        const int rr = i >> 7, cc = (i & 127) * 2;
        *(v2f*)(&sa[rr][cc]) = *(const v2f*)(x_act + (size_t)(tok0 + rr) * DI + cc);
    }
    __syncthreads();

    const int wave = tid >> 5;
    const int lane = tid & 31;
    const int half = lane >> 4;
    const int r    = lane & 15;

    for (int tile = wave; tile < 36; tile += 8) {
        const int ct = tile % 18;
        const int rt = tile / 18;
        const int n  = ct * 16 + r;                    // 0..287, one stream
        const float* bcol = BT + n + (size_t)(2 * half) * NB;
        v8f acc = {0.f, 0.f, 0.f, 0.f, 0.f, 0.f, 0.f, 0.f};
#pragma unroll 8
        for (int k0 = 0; k0 < DI; k0 += 4) {
            const v2f a = *(const v2f*)(&sa[rt * 16 + r][k0 + 2 * half]); // ds b64
            v2f b;
            b[0] = bcol[(size_t)k0 * NB];              // unconditional, coalesced
            b[1] = bcol[(size_t)k0 * NB + NB];
            acc = wmma_f32_16x16x4(a, b, acc);
        }
        if (ct < 16) {                                  // delta epilogue
            const float bias = dt_b[n];
#pragma unroll
            for (int v = 0; v < 8; ++v) {
                const int m = rt * 16 + v + half * 8;
                const float xv = acc[v] + bias;
                const float sp = (xv > 20.f) ? xv : log1pf(__expf(xv));  // softplus
                v2f pair;
                pair[0] = sp;
                pair[1] = sp * sa[m][n];          // delta * u
                *(v2f*)(dd + (size_t)(tok0 + m) * (2 * DI) + 2 * n) = pair;
            }
        } else {                                        // B/C epilogue
            const int nb = n - 256;                     // 0..31
#pragma unroll
            for (int v = 0; v < 8; ++v) {
                const int m = rt * 16 + v + half * 8;
                if (nb < DS) Bm[(size_t)(tok0 + m) * DS + nb] = acc[v];
                else         Cm[(size_t)(tok0 + m) * DS + (nb - DS)] = acc[v];
            }
        }
    }
}

// ---------------- K4: selective scan (final state only) ---------------------
// grid = 8 (batch), block = 1024: 4 lanes per channel d, 4 states each.
__global__ __launch_bounds__(1024)
void scan_kernel(const float* __restrict__ dd,      // (BL,256,2)
                 const float* __restrict__ x_act,   // (BL,256) last token only
                 const float* __restrict__ Bm,
                 const float* __restrict__ Cm,
                 const float* __restrict__ A_log,   // (256,16)
                 const float* __restrict__ Dp,      // (256,)
                 const float* __restrict__ z_last,  // (8,256)
                 float* __restrict__ y_pre) {       // (8,256)
    const int b   = blockIdx.x;
    const int tid = threadIdx.x;
    const int d   = tid >> 2;
    const int q   = tid & 3;
    const float LOG2E = 1.4426950408889634f;
    float Ar[4];
#pragma unroll
    for (int n = 0; n < 4; ++n)
        Ar[n] = -__expf(A_log[d * DS + q * 4 + n]) * LOG2E;  // A*log2e, A=-exp(A_log)
    float h[4] = {0.f, 0.f, 0.f, 0.f};

    __shared__ float sB[64][16];
    const size_t base = (size_t)b * SEQ;
    const float* ddp = dd + base * (2 * DI) + 2 * d;
    for (int t0 = 0; t0 < SEQ; t0 += 64) {
        __syncthreads();
        sB[tid >> 4][tid & 15] = Bm[(base + t0 + (tid >> 4)) * DS + (tid & 15)];
        __syncthreads();
        for (int tt = 0; tt < 64; ++tt) {
            const v2f dv = *(const v2f*)(ddp + (size_t)(t0 + tt) * (2 * DI));
            const float dlt = dv[0], du = dv[1];
            const v4f bb = *(const v4f*)(&sB[tt][q * 4]);     // one b128 LDS read
#pragma unroll
            for (int n = 0; n < 4; ++n)
                h[n] = exp2f(dlt * Ar[n]) * h[n] + bb[n] * du;
        }
    }
    float y = 0.f;
#pragma unroll
    for (int n = 0; n < 4; ++n) y += h[n] * Cm[(base + SEQ - 1) * DS + q * 4 + n];
    y += __shfl_xor(y, 1, 32);
    y += __shfl_xor(y, 2, 32);
    if (q == 0) {
        const float u = x_act[(base + SEQ - 1) * DI + d];
        y += Dp[d] * u;
        const float z = z_last[b * DI + d];
        y *= z / (1.f + __expf(-z));   // * silu(z)
        y_pre[b * DI + d] = y;
    }
}

// ---------------- K5: per-branch out_proj at last token ---------------------
__global__ void outproj_kernel(const float* __restrict__ h_last,   // (8,128)
                               const float* __restrict__ y_pre,    // (8,256)
                               const float* __restrict__ out_proj, // (128,256)
                               float* __restrict__ z_br) {         // (8,128)
    const int idx = blockIdx.x * blockDim.x + threadIdx.x;  // 0..1023
    const int bb = idx >> 7, m = idx & 127;
    float s = h_last[bb * DM + m];
    const float* w = out_proj + (size_t)m * DI;
    const float* y = y_pre + bb * DI;
#pragma unroll 8
    for (int k = 0; k < DI; ++k) s += y[k] * w[k];
    z_br[idx] = s;
}

// ---------------- K6: fusion GEMM + LN + classifier -------------------------
__global__ void fusion_kernel(const float* __restrict__ z_br,  // (3,8,128)
                              const float* __restrict__ W_fp,  // (128,384)
                              const float* __restrict__ b_fp,
                              const float* __restrict__ ln_g,
                              const float* __restrict__ ln_b,
                              const float* __restrict__ W_cls, // (1,128)
                              const float* __restrict__ b_cls,
                              float* __restrict__ out) {       // (8,1)
    __shared__ float f[8][128];
    const int tid = threadIdx.x;  // 256
    for (int idx = tid; idx < 1024; idx += 256) {
        const int bb = idx >> 7, j = idx & 127;
        float s = b_fp[j];
        const float* w = W_fp + (size_t)j * 384;
        for (int c = 0; c < 384; ++c) {
            const float zc = z_br[(c >> 7) * 1024 + bb * 128 + (c & 127)];
            s += zc * w[c];
        }
        f[bb][j] = s;
    }
    __syncthreads();
    {   // LayerNorm: one wave per row
        const int wv = tid >> 5, ln = tid & 31;
        float s = 0.f, s2 = 0.f;
#pragma unroll
        for (int i = 0; i < 4; ++i) {
            const float v = f[wv][ln * 4 + i];
            s += v; s2 += v * v;
        }
#pragma unroll
        for (int m = 16; m >= 1; m >>= 1) {
            s  += __shfl_xor(s,  m, 32);
            s2 += __shfl_xor(s2, m, 32);
        }
        const float mean = s * (1.f / 128.f);
        const float var  = s2 * (1.f / 128.f) - mean * mean;
        const float rstd = rsqrtf(var + 1e-5f);
#pragma unroll
        for (int i = 0; i < 4; ++i) {
            const int j = ln * 4 + i;
            f[wv][j] = (f[wv][j] - mean) * rstd * ln_g[j] + ln_b[j];
        }
    }
    __syncthreads();
    if (tid < 8) {
        float s = b_cls[0];
        for (int j = 0; j < 128; ++j) s += f[tid][j] * W_cls[j];
        out[tid] = s;
    }
}

// ---------------------------------------------------------------------------
extern "C" void kernel_launch(void* const* d_in, const int* in_sizes, int n_in,
                              void* d_out, int out_size, void* d_ws, size_t ws_size,
                              hipStream_t stream) {
    (void)in_sizes; (void)n_in; (void)out_size; (void)ws_size;
    // Workspace layout (floats), reused across the 3 branches:
    float* ws = (float*)d_ws;
    float* x_in    = ws;                        // BL*256
    float* x_act   = x_in   + (size_t)BL * DI;  // BL*256
    float* dd      = x_act  + (size_t)BL * DI;  // BL*256*2 (delta, delta*u)
    float* Bm      = dd     + (size_t)BL * DI * 2;  // BL*16
    float* Cm      = Bm     + (size_t)BL * DS;  // BL*16
    float* BT      = Cm     + (size_t)BL * DS;  // 256*288
    float* ipT     = BT     + DI * NB;          // 128*256
    float* h_last  = ipT    + DM * DI;          // 8*128
    float* hn_last = h_last + BATCH * DM;       // 8*128
    float* z_last  = hn_last + BATCH * DM;      // 8*256
    float* y_pre   = z_last + BATCH * DI;       // 8*256
    float* z_br    = y_pre  + BATCH * DI;       // 3*8*128 (persistent)

    for (int br = 0; br < 3; ++br) {
        const float* x = (const float*)d_in[br];
        const int pi = 3 + 13 * br;
        const float* W_emb    = (const float*)d_in[pi + 0];
        const float* b_emb    = (const float*)d_in[pi + 1];
        const float* ln_g     = (const float*)d_in[pi + 2];
        const float* ln_b     = (const float*)d_in[pi + 3];
        const float* in_proj  = (const float*)d_in[pi + 4];
        const float* conv_w   = (const float*)d_in[pi + 5];
        const float* conv_b   = (const float*)d_in[pi + 6];
        const float* x_proj   = (const float*)d_in[pi + 7];
        const float* dt_w     = (const float*)d_in[pi + 8];
        const float* dt_b     = (const float*)d_in[pi + 9];
        const float* A_log    = (const float*)d_in[pi + 10];
        const float* Dp       = (const float*)d_in[pi + 11];
        const float* out_proj = (const float*)d_in[pi + 12];

        transpose_ip_kernel<<<DM, DI, 0, stream>>>(in_proj, ipT);
        wprep_kernel<<<DI, NB, 0, stream>>>(dt_w, x_proj, BT);
        if (br == 0)
            branch_front_kernel<64><<<BL / 32, 256, 0, stream>>>(
                x, W_emb, b_emb, ln_g, ln_b, ipT, x_in, h_last, hn_last);
        else
            branch_front_kernel<128><<<BL / 32, 256, 0, stream>>>(
                x, W_emb, b_emb, ln_g, ln_b, ipT, x_in, h_last, hn_last);
        z_last_kernel<<<BATCH, DI, 0, stream>>>(hn_last, in_proj, z_last);
        conv_silu_kernel<<<(BL * DI) / 256, 256, 0, stream>>>(x_in, conv_w, conv_b, x_act);
        dbc_kernel<<<BL / 32, 256, 0, stream>>>(x_act, BT, dt_b, dd, Bm, Cm);
        scan_kernel<<<BATCH, 1024, 0, stream>>>(dd, x_act, Bm, Cm, A_log, Dp,
                                                z_last, y_pre);
        outproj_kernel<<<4, 256, 0, stream>>>(h_last, y_pre, out_proj,
                                              z_br + br * BATCH * DM);
    }

    const float* W_fp  = (const float*)d_in[42];
    const float* b_fp  = (const float*)d_in[43];
    const float* f_lng = (const float*)d_in[44];
    const float* f_lnb = (const float*)d_in[45];
    const float* W_cls = (const float*)d_in[46];
    const float* b_cls = (const float*)d_in[47];
    fusion_kernel<<<1, 256, 0, stream>>>(z_br, W_fp, b_fp, f_lng, f_lnb,
                                         W_cls, b_cls, (float*)d_out);
}